// GELU261_23648089932101
// MI455X (gfx1250) — compile-verified
//
#include <hip/hip_runtime.h>
#include <hip/hip_bf16.h>
#include <math.h>

// Problem constants (match reference)
#define BB 4
#define TT 4096
#define DD 2048
#define NN 512
#define RR (BB * TT)          // 16384 rows
#define FIRE_THRESH 0.85f
#define FACIL_RATE  2.0f
#define MAX_GATE    8.0f
#define NORM_EPS    1e-12f

// Workspace layout (float offsets). All regions 16B aligned.
#define WS_COL    0                       // [DD]  column sums of gelu(x)
#define WS_ROWSQ  (WS_COL + DD)           // [RR]  per-row sum of squares
#define WS_MN     (WS_ROWSQ + RR)         // [DD]  normalized mean vector
#define WS_SIMS   (WS_MN + DD)            // [NN]  similarities
#define WS_GATED  (WS_SIMS + NN)          // [DD]  per-channel gate
#define WS_MVN    (WS_GATED + DD)         // [DD]  normalized selected key
#define WS_TOTAL  (WS_MVN + DD)
#define WS_ZERO_COUNT (DD + RR)           // only colsum + rowsq are accumulators

typedef __attribute__((ext_vector_type(2))) float v2f;
typedef __attribute__((ext_vector_type(8))) float v8f;

__device__ __forceinline__ float gelu_tanh(float x) {
    const float c = 0.7978845608028654f; // sqrt(2/pi)
    float x3 = x * x * x;
    return 0.5f * x * (1.0f + tanhf(c * (x + 0.044715f * x3)));
}

__device__ __forceinline__ float waveReduceSum(float v) {
#pragma unroll
    for (int o = 16; o > 0; o >>= 1) v += __shfl_down(v, o, 32);
    return v;
}

__device__ __forceinline__ float blockReduceSum(float v, float* sh32) {
    const int lane = threadIdx.x & 31;
    const int w    = threadIdx.x >> 5;
    v = waveReduceSum(v);
    if (lane == 0) sh32[w] = v;
    __syncthreads();
    const int nw = blockDim.x >> 5;
    float r = (threadIdx.x < (unsigned)nw) ? sh32[threadIdx.x] : 0.0f;
    if (w == 0) {
        r = waveReduceSum(r);
        if (lane == 0) sh32[0] = r;
    }
    __syncthreads();
    r = sh32[0];
    __syncthreads();
    return r;
}

// ---------------- K0: zero accumulator regions ----------------
__global__ void zero_ws(float* ws, int n) {
    int i = blockIdx.x * blockDim.x + threadIdx.x;
    if (i < n) ws[i] = 0.0f;
}

// ---------------- KA: y = gelu(x); colsum; row sumsq ----------------
// grid = 512 blocks * 256 threads; each block handles 32 rows;
// thread t owns columns [8t, 8t+8).
__global__ void __launch_bounds__(256) gelu_stats(const float* __restrict__ x,
                                                  float* __restrict__ y,
                                                  float* __restrict__ colsum,
                                                  float* __restrict__ rowsq) {
    const int t  = threadIdx.x;
    const int c0 = t * 8;
    const int r0 = blockIdx.x * 32;
    float acc[8];
#pragma unroll
    for (int j = 0; j < 8; ++j) acc[j] = 0.0f;

    for (int r = 0; r < 32; ++r) {
        const int row = r0 + r;
        const size_t base = (size_t)row * DD + c0;
        const float4 a = *(const float4*)(x + base);
        const float4 b = *(const float4*)(x + base + 4);
        float v[8] = {a.x, a.y, a.z, a.w, b.x, b.y, b.z, b.w};
        float sq = 0.0f;
#pragma unroll
        for (int j = 0; j < 8; ++j) {
            v[j] = gelu_tanh(v[j]);
            acc[j] += v[j];
            sq += v[j] * v[j];
        }
        float4 o0 = {v[0], v[1], v[2], v[3]};
        float4 o1 = {v[4], v[5], v[6], v[7]};
        *(float4*)(y + base)     = o0;
        *(float4*)(y + base + 4) = o1;
        sq = waveReduceSum(sq);
        if ((t & 31) == 0) atomicAdd(&rowsq[row], sq);
    }
#pragma unroll
    for (int j = 0; j < 8; ++j) atomicAdd(&colsum[c0 + j], acc[j]);
}

// ---------------- K1: m_n = normalize(colsum / R) ----------------
__global__ void __launch_bounds__(256) mean_norm(const float* __restrict__ colsum,
                                                 float* __restrict__ mn) {
    __shared__ float sh[32];
    const int t = threadIdx.x;
    float vals[8];
    float s = 0.0f;
#pragma unroll
    for (int j = 0; j < 8; ++j) {
        vals[j] = colsum[t * 8 + j] * (1.0f / (float)RR);
        s += vals[j] * vals[j];
    }
    const float n2  = blockReduceSum(s, sh);
    const float inv = 1.0f / fmaxf(sqrtf(n2), NORM_EPS);
#pragma unroll
    for (int j = 0; j < 8; ++j) mn[t * 8 + j] = vals[j] * inv;
}

// ---------------- K2: sims[n] = mask ? keys_n[n] . m_n : -1 ----------------
__global__ void __launch_bounds__(256) sims_kernel(const float* __restrict__ keys,
                                                   const float* __restrict__ mn,
                                                   const unsigned char* __restrict__ mask,
                                                   float* __restrict__ sims) {
    __shared__ float sh[32];
    const int n = blockIdx.x;
    const int t = threadIdx.x;
    const float* kr = keys + (size_t)n * DD;
    float dot = 0.0f, ksq = 0.0f;
    for (int i = t; i < DD; i += 256) {
        const float kv = kr[i];
        dot += kv * mn[i];
        ksq += kv * kv;
    }
    dot = blockReduceSum(dot, sh);
    ksq = blockReduceSum(ksq, sh);
    if (t == 0) {
        const float s = dot / fmaxf(sqrtf(ksq), NORM_EPS);
        sims[n] = mask[n] ? s : -1.0f;
    }
}

// ---------------- K3: argmax -> gate_d, mv_n ----------------
__global__ void __launch_bounds__(256) select_gate(const float* __restrict__ sims,
                                                   const float* __restrict__ facil,
                                                   const float* __restrict__ mag,
                                                   const float* __restrict__ keys,
                                                   const float* __restrict__ log_k_mag,
                                                   float* __restrict__ gate_d,
                                                   float* __restrict__ mvn) {
    __shared__ float sv[256];
    __shared__ int   si[256];
    __shared__ float sh[32];
    const int t = threadIdx.x;

    // argmax (first-max tie-break, like jnp.argmax)
    float best = -1e30f;
    int   bi   = 0;
    for (int n = t; n < NN; n += 256) {
        const float s = sims[n];
        if (s > best) { best = s; bi = n; }
    }
    sv[t] = best; si[t] = bi;
    __syncthreads();
    for (int off = 128; off > 0; off >>= 1) {
        if (t < off) {
            if (sv[t + off] > sv[t] ||
                (sv[t + off] == sv[t] && si[t + off] < si[t])) {
                sv[t] = sv[t + off];
                si[t] = si[t + off];
            }
        }
        __syncthreads();
    }
    const int   idx  = si[0];
    const float smax = sv[0];
    __syncthreads();

    const float f0 = facil[idx];
    const float facil_level = (smax > FIRE_THRESH) ? f0 * FACIL_RATE : f0;

    const float* mrow = mag  + (size_t)idx * DD;
    const float* krow = keys + (size_t)idx * DD;

    float msum = 0.0f, ksq = 0.0f;
    for (int i = t; i < DD; i += 256) {
        msum += mrow[i];
        const float kv = krow[i];
        ksq += kv * kv;
    }
    msum = blockReduceSum(msum, sh);
    ksq  = blockReduceSum(ksq, sh);

    const float magmean = fmaxf(msum / (float)DD, 1e-6f);
    const float kinv    = 1.0f / fmaxf(sqrtf(ksq), NORM_EPS);
    const float kmag    = fminf(fmaxf(expf(log_k_mag[0]), 0.01f), 5.0f);
    const float scale   = kmag * (facil_level - 1.0f);

    for (int i = t; i < DD; i += 256) {
        const float imp = mrow[i] / magmean;
        gate_d[i] = fminf(1.0f + scale * imp, MAX_GATE);
        mvn[i]    = krow[i] * kinv;
    }
}

// ---------------- K4: WMMA f32 GEMV (tok_sim) + in-place gating ----------------
// One wave owns 16 rows. A tile = 16x4 f32 of y (lane l: row l&15, K-pair (l>>4)*2),
// B tile = mv_n chunk in column 0 (lanes 0/16), zero elsewhere.
// C column 0 accumulates the 16 dot products (lane 0: M=0..7, lane 16: M=8..15).
__global__ void __launch_bounds__(64) wmma_gate(float* __restrict__ y,
                                                const float* __restrict__ mvn,
                                                const float* __restrict__ gate_d,
                                                const float* __restrict__ rowsq) {
    const int lane    = threadIdx.x & 31;
    const int wave    = threadIdx.x >> 5;
    const int rowbase = (blockIdx.x * 2 + wave) * 16;
    const int arow    = rowbase + (lane & 15);
    const int coff    = (lane >> 4) << 1;          // 0 or 2
    const float* aptr = y + (size_t)arow * DD + coff;
    const bool   sel  = (lane & 15) == 0;          // lanes 0 and 16 carry column 0 of B

    v8f c = {};
#pragma unroll 4
    for (int kb = 0; kb < DD; kb += 4) {
        const float2 av = *(const float2*)(aptr + kb);
        const float2 mv = *(const float2*)(mvn + kb + coff);
        v2f a; a.x = av.x;              a.y = av.y;
        v2f b; b.x = sel ? mv.x : 0.0f; b.y = sel ? mv.y : 0.0f;
        c = __builtin_amdgcn_wmma_f32_16x16x4_f32(false, a, false, b,
                                                  (short)0, c, false, false);
    }

    // Broadcast the 16 dots (column 0 of C) to all lanes.
    float dots[16];
#pragma unroll
    for (int j = 0; j < 8; ++j) {
        dots[j]     = __shfl(c[j], 0, 32);
        dots[j + 8] = __shfl(c[j], 16, 32);
    }

    // Gate the 16x2048 tile in place (tile is hot in L2 after the WMMA pass).
    const float4* pg = (const float4*)gate_d;
    for (int r = 0; r < 16; ++r) {
        const int row = rowbase + r;
        float tok = dots[r] / fmaxf(sqrtf(rowsq[row]), NORM_EPS);
        tok = fminf(fmaxf(tok, 0.0f), 1.0f);
        const float omt = 1.0f - tok;
        float4* py = (float4*)(y + (size_t)row * DD);
        for (int i = lane; i < DD / 4; i += 32) {
            float4 yv = py[i];
            const float4 gv = pg[i];
            yv.x *= gv.x * tok + omt;
            yv.y *= gv.y * tok + omt;
            yv.z *= gv.z * tok + omt;
            yv.w *= gv.w * tok + omt;
            py[i] = yv;
        }
    }
}

extern "C" void kernel_launch(void* const* d_in, const int* in_sizes, int n_in,
                              void* d_out, int out_size, void* d_ws, size_t ws_size,
                              hipStream_t stream) {
    const float*         x     = (const float*)d_in[0];
    const float*         logk  = (const float*)d_in[1];
    const float*         keys  = (const float*)d_in[2];
    const float*         mag   = (const float*)d_in[3];
    const float*         facil = (const float*)d_in[4];
    const unsigned char* mask  = (const unsigned char*)d_in[5];

    float* y  = (float*)d_out;       // y buffer doubles as output (gated in place)
    float* ws = (float*)d_ws;

    float* colsum = ws + WS_COL;
    float* rowsq  = ws + WS_ROWSQ;
    float* mn     = ws + WS_MN;
    float* sims   = ws + WS_SIMS;
    float* gate_d = ws + WS_GATED;
    float* mvn    = ws + WS_MVN;

    // K0: zero the accumulator regions (colsum + rowsq) every launch.
    zero_ws<<<(WS_ZERO_COUNT + 255) / 256, 256, 0, stream>>>(ws, WS_ZERO_COUNT);

    // KA: gelu + column sums + row sum-of-squares. 512 blocks x 32 rows.
    gelu_stats<<<RR / 32, 256, 0, stream>>>(x, y, colsum, rowsq);

    // K1: normalized mean vector.
    mean_norm<<<1, 256, 0, stream>>>(colsum, mn);

    // K2: similarities per key.
    sims_kernel<<<NN, 256, 0, stream>>>(keys, mn, mask, sims);

    // K3: argmax + per-channel gate + normalized selected key.
    select_gate<<<1, 256, 0, stream>>>(sims, facil, mag, keys, logk, gate_d, mvn);

    // K4: WMMA f32 GEMV for tok_sim, fused with in-place gating.
    // 16384 rows / 16 per wave / 2 waves per block = 512 blocks.
    wmma_gate<<<RR / 32, 64, 0, stream>>>(y, mvn, gate_d, rowsq);
}